// SNRI_52475910423272
// MI455X (gfx1250) — compile-verified
//
#include <hip/hip_runtime.h>
#include <math.h>

#define NN 20000      // nodes
#define NE 400000     // edges
#define DD 128        // feature dim
#define NR 64         // relations
#define NB 4          // bases
#define TILE_M 64
#define MAX_TILES (((NE + TILE_M - 1) / TILE_M) + NR)   // 6314

typedef float v2f __attribute__((ext_vector_type(2)));
typedef float v8f __attribute__((ext_vector_type(8)));

// Packed fragment layout for a 128x128 B matrix:
//   Bp[f][n] = { B[2f][n], B[2f+1][n] },  f in [0,64), n in [0,128)
// At WMMA k-step kk, lane (hi,nl) of the wave owning columns [n0,n0+16) loads
// the single 8-byte word Bp[2*kk+hi][n0+nl]  -> one global_load_b64, coalesced.

// ---------------- small precompute kernels ----------------

__global__ void k_zero(int* cnt, int* wl_n) {
    int t = threadIdx.x;
    if (t < NR) cnt[t] = 0;
    if (t == NR) wl_n[0] = 0;
}

// Packed relation weights: Wp[r][f][n] = {W_r[2f][n], W_r[2f+1][n]},
// W_r[k][n] = sum_b w_comp[r][b] * weight[b][k][n]
__global__ void k_relweights(const float* __restrict__ w_comp,
                             const float* __restrict__ weight,
                             float* __restrict__ Wp) {
    int idx = blockIdx.x * blockDim.x + threadIdx.x;   // 64*64*128
    if (idx >= NR * 64 * DD) return;
    int r = idx >> 13;              // / (64*128)
    int f = (idx >> 7) & 63;
    int n = idx & (DD - 1);
    float s0 = 0.f, s1 = 0.f;
    #pragma unroll
    for (int b = 0; b < NB; ++b) {
        float c = w_comp[r * NB + b];
        s0 += c * weight[b * DD * DD + (2 * f) * DD + n];
        s1 += c * weight[b * DD * DD + (2 * f + 1) * DD + n];
    }
    v2f v; v.x = s0; v.y = s1;
    *(v2f*)&Wp[(size_t)idx * 2] = v;
}

// Repack a plain row-major 128x128 matrix into fragment-packed layout
__global__ void k_pack128(const float* __restrict__ B, float* __restrict__ Bp) {
    int idx = blockIdx.x * blockDim.x + threadIdx.x;   // 64*128
    if (idx >= 64 * DD) return;
    int f = idx >> 7, n = idx & (DD - 1);
    v2f v; v.x = B[(2 * f) * DD + n]; v.y = B[(2 * f + 1) * DD + n];
    *(v2f*)&Bp[(size_t)idx * 2] = v;
}

// T1[r][n] = attn_tab[r] @ A_w[256:288] + A_b ; T2[r][n] = attn_tab[r] @ A_w[288:320]
__global__ void k_attntabs(const float* __restrict__ attn_tab,
                           const float* __restrict__ A_w,
                           const float* __restrict__ A_b,
                           float* __restrict__ T1, float* __restrict__ T2) {
    int r = blockIdx.x, n = threadIdx.x;
    float s1 = A_b[n], s2 = 0.f;
    #pragma unroll
    for (int j = 0; j < 32; ++j) {
        float t = attn_tab[r * 32 + j];
        s1 += t * A_w[(256 + j) * DD + n];
        s2 += t * A_w[(288 + j) * DD + n];
    }
    T1[r * DD + n] = s1;
    T2[r * DD + n] = s2;
}

// rel_out = rel_emb @ w_rel, last row zeroed  (written straight to d_out tail)
__global__ void k_relout(const float* __restrict__ rel_emb,
                         const float* __restrict__ w_rel,
                         float* __restrict__ out) {
    int r = blockIdx.x, n = threadIdx.x;
    float s = 0.f;
    if (r != NR - 1)
        for (int k = 0; k < DD; ++k)
            s += rel_emb[r * DD + k] * w_rel[k * DD + n];
    out[r * DD + n] = s;
}

// ---------------- generic M x 128 @ 128 x 128 fp32 WMMA GEMM ----------------
// B in packed-fragment layout. block = 256 threads (8 waves); 64 rows per block.

__global__ __launch_bounds__(256) void k_gemm128(const float* __restrict__ A,
                                                 const float* __restrict__ Bp,
                                                 float* __restrict__ C, int M) {
    __shared__ float At[TILE_M][DD + 4];   // pad to 132 floats: no A-frag bank conflicts
    const int tid = threadIdx.x;
    const int wave = tid >> 5, lane = tid & 31;
    const int hi = lane >> 4, nl = lane & 15;
    const int row0 = blockIdx.x * TILE_M;
    const int n0 = wave * 16;
    const bool full = (row0 + TILE_M <= M);

    // prefetch this wave's B slice while we stage A
    for (int f = lane; f < 64; f += 32)
        __builtin_prefetch(&Bp[(size_t)f * 256 + n0 * 2], 0, 3);

    // stage A tile (zero-pad OOB rows)
    if (full) {
        for (int i = tid; i < TILE_M * 32; i += 256) {
            int m = i >> 5, c4 = (i & 31) * 4;
            float4 v = *(const float4*)&A[(size_t)(row0 + m) * DD + c4];
            At[m][c4 + 0] = v.x; At[m][c4 + 1] = v.y;
            At[m][c4 + 2] = v.z; At[m][c4 + 3] = v.w;
        }
    } else {
        for (int i = tid; i < TILE_M * 32; i += 256) {
            int m = i >> 5, c4 = (i & 31) * 4;
            float4 v = make_float4(0.f, 0.f, 0.f, 0.f);
            if (row0 + m < M) v = *(const float4*)&A[(size_t)(row0 + m) * DD + c4];
            At[m][c4 + 0] = v.x; At[m][c4 + 1] = v.y;
            At[m][c4 + 2] = v.z; At[m][c4 + 3] = v.w;
        }
    }
    __syncthreads();

    v8f acc[4] = {{}, {}, {}, {}};
    for (int kk = 0; kk < 32; ++kk) {
        int kb = kk * 4 + 2 * hi;                    // lanes 0-15: K{0,1}; 16-31: K{2,3}
        v2f bf = *(const v2f*)&Bp[(size_t)(2 * kk + hi) * 256 + (n0 + nl) * 2];
        #pragma unroll
        for (int mt = 0; mt < 4; ++mt) {
            v2f af; af.x = At[mt * 16 + nl][kb];
            af.y = At[mt * 16 + nl][kb + 1];
            acc[mt] = __builtin_amdgcn_wmma_f32_16x16x4_f32(
                false, af, false, bf, (short)0, acc[mt], false, false);
        }
    }
    if (full) {
        #pragma unroll
        for (int mt = 0; mt < 4; ++mt)
            #pragma unroll
            for (int v = 0; v < 8; ++v) {
                int m = row0 + mt * 16 + v + 8 * hi;
                C[(size_t)m * DD + n0 + nl] = acc[mt][v];
            }
    } else {
        #pragma unroll
        for (int mt = 0; mt < 4; ++mt)
            #pragma unroll
            for (int v = 0; v < 8; ++v) {
                int m = row0 + mt * 16 + v + 8 * hi;
                if (m < M) C[(size_t)m * DD + n0 + nl] = acc[mt][v];
            }
    }
}

// ---------------- edge binning ----------------

__global__ void k_hist(const int* __restrict__ etype, int* __restrict__ cnt) {
    int e = blockIdx.x * blockDim.x + threadIdx.x;
    if (e < NE) atomicAdd(&cnt[etype[e]], 1);
}

__global__ void k_scan(const int* __restrict__ cnt, int* __restrict__ off,
                       int* __restrict__ cur, int* __restrict__ wl,
                       int* __restrict__ wl_n) {
    if (threadIdx.x != 0) return;
    int s = 0, w = 0;
    for (int r = 0; r < NR; ++r) {
        off[r] = s; cur[r] = s;
        int c = cnt[r]; s += c;
        int nt = (c + TILE_M - 1) / TILE_M;
        for (int t = 0; t < nt; ++t) wl[w++] = (r << 16) | t;
    }
    wl_n[0] = w;
}

__global__ void k_scatter(const int* __restrict__ etype, int* __restrict__ cur,
                          int* __restrict__ perm) {
    int e = blockIdx.x * blockDim.x + threadIdx.x;
    if (e < NE) perm[atomicAdd(&cur[etype[e]], 1)] = e;
}

// ---------------- main edge kernel: WMMA msg GEMM + attention + atomic scatter --------
// block = 256 threads (8 waves), one 64-edge same-relation tile per block

__global__ __launch_bounds__(256) void k_edges(
    const float* __restrict__ h, const float* __restrict__ rel_emb,
    const float* __restrict__ Wp_all, const float* __restrict__ T1,
    const float* __restrict__ T2, const float* __restrict__ HA1,
    const float* __restrict__ HA2, const float* __restrict__ B_w,
    const float* __restrict__ B_b, const int* __restrict__ src,
    const int* __restrict__ dst, const int* __restrict__ elabel,
    const int* __restrict__ perm, const int* __restrict__ off,
    const int* __restrict__ cnt, const int* __restrict__ wl,
    const int* __restrict__ wl_n, float* __restrict__ accum) {

    __shared__ float At[TILE_M][DD + 4];
    __shared__ float relrow[DD];
    __shared__ float a_s[TILE_M];
    __shared__ int   s_s[TILE_M], d_s[TILE_M], l_s[TILE_M];

    const int b = blockIdx.x;
    if (b >= wl_n[0]) return;
    const int ent = wl[b];
    const int r = ent >> 16, tile = ent & 0xFFFF;
    const int base = off[r] + tile * TILE_M;
    const int count = min(TILE_M, cnt[r] - tile * TILE_M);
    if (count <= 0) return;

    const int tid = threadIdx.x;
    const int wave = tid >> 5, lane = tid & 31;
    const int hi = lane >> 4, nl = lane & 15;
    const int n0 = wave * 16;
    const float* Wp = Wp_all + (size_t)r * 64 * 256;   // this relation's packed W

    // prefetch this wave's W slice; lines arrive while we stage A + do attention
    for (int f = lane; f < 64; f += 32)
        __builtin_prefetch(&Wp[(size_t)f * 256 + n0 * 2], 0, 3);

    if (tid < DD) relrow[tid] = rel_emb[r * DD + tid];
    if (tid < TILE_M) {
        if (tid < count) {
            int e = perm[base + tid];
            s_s[tid] = src[e]; d_s[tid] = dst[e]; l_s[tid] = elabel[e];
        } else { s_s[tid] = 0; d_s[tid] = 0; l_s[tid] = 0; }
    }
    __syncthreads();

    // stage edge_data = h[src] * rel_emb[r]
    for (int i = tid; i < TILE_M * 32; i += 256) {
        int m = i >> 5, c4 = (i & 31) * 4;
        float4 v = make_float4(0.f, 0.f, 0.f, 0.f);
        if (m < count) {
            float4 hv = *(const float4*)&h[(size_t)s_s[m] * DD + c4];
            v = make_float4(hv.x * relrow[c4], hv.y * relrow[c4 + 1],
                            hv.z * relrow[c4 + 2], hv.w * relrow[c4 + 3]);
        }
        At[m][c4 + 0] = v.x; At[m][c4 + 1] = v.y;
        At[m][c4 + 2] = v.z; At[m][c4 + 3] = v.w;
    }

    // attention: wave w handles edges [8w, 8w+8); lanes split the 128-dim dot
    const float bb0 = B_b[0];
    for (int i = 0; i < 8; ++i) {
        int m = wave * 8 + i;
        if (m < count) {
            int s = s_s[m], d = d_s[m], l = l_s[m];
            float acc = 0.f;
            for (int c = lane; c < DD; c += 32) {
                float pre = HA1[(size_t)s * DD + c] + HA2[(size_t)d * DD + c] +
                            T1[r * DD + c] + T2[l * DD + c];
                acc += fmaxf(pre, 0.f) * B_w[c];
            }
            for (int o = 16; o > 0; o >>= 1) acc += __shfl_down(acc, o, 32);
            if (lane == 0) a_s[m] = 1.0f / (1.0f + __expf(-(acc + bb0)));
        }
    }
    __syncthreads();

    // msg = edge_data @ W_r  (fp32 WMMA, K=128 in 32 steps)
    v8f acc[4] = {{}, {}, {}, {}};
    for (int kk = 0; kk < 32; ++kk) {
        int kb = kk * 4 + 2 * hi;
        v2f bf = *(const v2f*)&Wp[(size_t)(2 * kk + hi) * 256 + (n0 + nl) * 2];
        #pragma unroll
        for (int mt = 0; mt < 4; ++mt) {
            v2f af; af.x = At[mt * 16 + nl][kb];
            af.y = At[mt * 16 + nl][kb + 1];
            acc[mt] = __builtin_amdgcn_wmma_f32_16x16x4_f32(
                false, af, false, bf, (short)0, acc[mt], false, false);
        }
    }

    // scatter a * msg into accumulator (lanes 0-15 of a frag row hit 16 consecutive n)
    if (count == TILE_M) {
        #pragma unroll
        for (int mt = 0; mt < 4; ++mt)
            #pragma unroll
            for (int v = 0; v < 8; ++v) {
                int m = mt * 16 + v + 8 * hi;
                atomicAdd(&accum[(size_t)d_s[m] * DD + n0 + nl], a_s[m] * acc[mt][v]);
            }
    } else {
        #pragma unroll
        for (int mt = 0; mt < 4; ++mt)
            #pragma unroll
            for (int v = 0; v < 8; ++v) {
                int m = mt * 16 + v + 8 * hi;
                if (m < count)
                    atomicAdd(&accum[(size_t)d_s[m] * DD + n0 + nl], a_s[m] * acc[mt][v]);
            }
    }
}

__global__ void k_relu(float* __restrict__ x, int n) {
    int i = blockIdx.x * blockDim.x + threadIdx.x;
    if (i < n) x[i] = fmaxf(x[i], 0.f);
}

// ---------------- launch ----------------

extern "C" void kernel_launch(void* const* d_in, const int* in_sizes, int n_in,
                              void* d_out, int out_size, void* d_ws, size_t ws_size,
                              hipStream_t stream) {
    const float* h       = (const float*)d_in[0];
    const float* rel_emb = (const float*)d_in[1];
    const float* attn    = (const float*)d_in[2];
    const float* weight  = (const float*)d_in[3];
    const float* w_comp  = (const float*)d_in[4];
    const float* w_rel   = (const float*)d_in[5];
    const float* slw     = (const float*)d_in[6];
    const float* A_w     = (const float*)d_in[7];
    const float* A_b     = (const float*)d_in[8];
    const float* B_w     = (const float*)d_in[9];
    const float* B_b     = (const float*)d_in[10];
    const int*   src     = (const int*)d_in[11];
    const int*   dst     = (const int*)d_in[12];
    const int*   etype   = (const int*)d_in[13];
    const int*   elabel  = (const int*)d_in[14];

    float* out = (float*)d_out;                 // [20000*128] node_repr, then [64*128] rel_out

    // workspace carve (~26.6 MB)
    float* Wp   = (float*)d_ws;                  // packed relation weights: 64*64*256
    float* T1   = Wp + (size_t)NR * 64 * 256;    // 64*128
    float* T2   = T1 + NR * DD;
    float* HA1  = T2 + NR * DD;                  // 20000*128
    float* HA2  = HA1 + (size_t)NN * DD;
    float* A1p  = HA2 + (size_t)NN * DD;         // packed A_w[0:128]   : 64*256
    float* A2p  = A1p + 64 * 256;                // packed A_w[128:256]
    float* SLp  = A2p + 64 * 256;                // packed slw
    int*   cnt  = (int*)(SLp + 64 * 256);        // 64
    int*   offs = cnt + NR;
    int*   cur  = offs + NR;
    int*   wl_n = cur + NR;                      // 1 (padded)
    int*   wl   = wl_n + 64;                     // <= 6314 entries
    int*   perm = wl + 8192;                     // 400000

    k_zero<<<1, 256, 0, stream>>>(cnt, wl_n);
    k_relweights<<<(NR * 64 * DD + 255) / 256, 256, 0, stream>>>(w_comp, weight, Wp);
    k_pack128<<<32, 256, 0, stream>>>(A_w, A1p);
    k_pack128<<<32, 256, 0, stream>>>(A_w + DD * DD, A2p);
    k_pack128<<<32, 256, 0, stream>>>(slw, SLp);
    k_attntabs<<<NR, DD, 0, stream>>>(attn, A_w, A_b, T1, T2);
    k_relout<<<NR, DD, 0, stream>>>(rel_emb, w_rel, out + (size_t)NN * DD);

    const int gblk = (NN + TILE_M - 1) / TILE_M;            // 313
    k_gemm128<<<gblk, 256, 0, stream>>>(h, A1p, HA1, NN);   // h @ A_w[0:128]
    k_gemm128<<<gblk, 256, 0, stream>>>(h, A2p, HA2, NN);   // h @ A_w[128:256]
    k_gemm128<<<gblk, 256, 0, stream>>>(h, SLp, out, NN);   // accumulator base h @ slw

    k_hist<<<(NE + 255) / 256, 256, 0, stream>>>(etype, cnt);
    k_scan<<<1, 64, 0, stream>>>(cnt, offs, cur, wl, wl_n);
    k_scatter<<<(NE + 255) / 256, 256, 0, stream>>>(etype, cur, perm);

    k_edges<<<MAX_TILES, 256, 0, stream>>>(h, rel_emb, Wp, T1, T2, HA1, HA2,
                                           B_w, B_b, src, dst, elabel, perm,
                                           offs, cnt, wl, wl_n, out);

    k_relu<<<(NN * DD + 255) / 256, 256, 0, stream>>>(out, NN * DD);
}